// rosa_emb_layer_7928509628834
// MI455X (gfx1250) — compile-verified
//
#include <hip/hip_runtime.h>
#include <stdint.h>

// Problem constants (from reference): B=8, T=2048, V=50257, C=768.
#define ROSA_B 8
#define ROSA_T 2048
#define ROSA_C 768

// ---------------------------------------------------------------------------
// Kernel 0: zero the per-(b,i) best-key array (workspace). 64 KB, every call.
// ---------------------------------------------------------------------------
__global__ __launch_bounds__(256) void rosa_init_kernel(int* __restrict__ best) {
    best[blockIdx.x * 256 + threadIdx.x] = 0;
}

// ---------------------------------------------------------------------------
// Kernel 1: diagonal run-length scan, wave-parallel (exact).
//   Along diagonal lag d:  e[k] = (x[k+d]==x[k]),
//   run length L[k] = k - max{ k' <= k : e[k']==0 }  (sentinel -1)
//   -> inclusive prefix-max of "zero positions" = wave shuffle scan + carry.
//   Winner key per (b,i): (L<<11) | (2047-d)   [max L, then smaller d = larger j,
//   matching the reference's j*1e-5 tie-break]. atomicMax pushes are rare
//   (~T^2/2V ~ 42 per batch for random tokens), so contention is negligible.
//   Grid: one wave per diagonal; 8 waves (256 thr) per block; 256 blocks/batch.
// ---------------------------------------------------------------------------
__global__ __launch_bounds__(256) void rosa_scan_kernel(const int* __restrict__ idx,
                                                        int* __restrict__ best) {
    __shared__ int xs[ROSA_T];
    const int b    = blockIdx.x >> 8;       // 256 blocks per batch
    const int blk  = blockIdx.x & 255;
    const int t    = threadIdx.x;
    const int lane = t & 31;
    const int wave = t >> 5;

    const int* xrow = idx + (b << 11);
    for (int k = t; k < ROSA_T; k += 256) xs[k] = xrow[k];
    __syncthreads();

    const int d = (blk << 3) + wave + 1;    // lag in [1, 2048]; 2048 invalid
    if (d >= ROSA_T) return;                // uniform per wave

    int* bb = best + (b << 11);
    const int n = ROSA_T - d;               // diagonal length
    int carry = -1;                         // last zero-position seen so far

    for (int base = 0; base < n; base += 32) {
        const int  k  = base + lane;
        const bool e  = (k < n) && (xs[k + d] == xs[k]);
        int z = e ? -1 : k;                 // zero-position, or neutral (-1)

        // inclusive prefix-max across the wave (5 shuffle steps)
        #pragma unroll
        for (int off = 1; off < 32; off <<= 1) {
            const int up = __shfl_up(z, off, 32);
            if (lane >= off && up > z) z = up;
        }

        if (e) {
            const int m = (z > carry) ? z : carry;
            const int L = k - m;            // exact run length ending at k
            atomicMax(&bb[k + d], (L << 11) | (2047 - d));
        }

        const int last = __shfl(z, 31, 32); // chunk-wide max zero-position
        if (last > carry) carry = last;
    }
}

// ---------------------------------------------------------------------------
// Kernel 2: decode best key + gather embedding row (3 KB) per (b,i).
//   Streams each row global->LDS->global via CDNA5 async LDS ops (ASYNCcnt).
//   Grid: B*T blocks of 192 threads; one float4 (16 B) per thread.
// ---------------------------------------------------------------------------
__global__ __launch_bounds__(192) void rosa_gather_kernel(const int* __restrict__ idx,
                                                          const float* __restrict__ emb,
                                                          const int* __restrict__ best,
                                                          float* __restrict__ out) {
    __shared__ float4 stage[192];
    const int r = blockIdx.x;               // r = b*T + i
    const int b = r >> 11;
    const int i = r & (ROSA_T - 1);
    const int t = threadIdx.x;

    float4* orow = reinterpret_cast<float4*>(out + (size_t)r * ROSA_C) + t;

    const int key = best[r];
    if ((key >> 11) == 0) {                 // no match -> zero row
        *orow = make_float4(0.f, 0.f, 0.f, 0.f);
        return;
    }

    const int d = 2047 - (key & 2047);      // winning lag
    int j = i - d + 1;                      // pred_idx = min(best_j+1, T-1)
    if (j > ROSA_T - 1) j = ROSA_T - 1;
    const int p = idx[(b << 11) + j];       // predicted token id

    const float4* grow = reinterpret_cast<const float4*>(emb + (size_t)p * ROSA_C) + t;

    // Per-lane private 16B LDS slot; low 32 bits of the flat shared pointer
    // are the wave-relative LDS byte address.
    const unsigned lds_off = (unsigned)(uintptr_t)&stage[t];
    const unsigned long long gsrc = (unsigned long long)(uintptr_t)grow;
    const unsigned long long gdst = (unsigned long long)(uintptr_t)orow;

    // global -> LDS (ASYNCcnt++)
    asm volatile("global_load_async_to_lds_b128 %0, %1, off"
                 :: "v"(lds_off), "v"(gsrc) : "memory");
    // Async ops are unordered w.r.t. LDS access: drain before the store reads LDS.
    asm volatile("s_wait_asynccnt 0x0" ::: "memory");
    // LDS -> global (ASYNCcnt++); S_ENDPGM's implicit wait-idle drains it.
    asm volatile("global_store_async_from_lds_b128 %0, %1, off"
                 :: "v"(gdst), "v"(lds_off) : "memory");
}

// ---------------------------------------------------------------------------
// Launch: init(best) -> wave-parallel diagonal scan -> async gather.
// All on `stream`, graph-safe. Workspace usage: B*T ints = 64 KB.
// ---------------------------------------------------------------------------
extern "C" void kernel_launch(void* const* d_in, const int* in_sizes, int n_in,
                              void* d_out, int out_size, void* d_ws, size_t ws_size,
                              hipStream_t stream) {
    const int*   idx  = (const int*)d_in[0];   // (B,T) token ids
    const float* emb  = (const float*)d_in[1]; // (V,C) fp32 table
    float*       out  = (float*)d_out;         // (B,T,C) fp32
    int*         best = (int*)d_ws;            // B*T best keys

    (void)in_sizes; (void)n_in; (void)out_size; (void)ws_size;

    rosa_init_kernel<<<(ROSA_B * ROSA_T) / 256, 256, 0, stream>>>(best);
    rosa_scan_kernel<<<ROSA_B * 256, 256, 0, stream>>>(idx, best);
    rosa_gather_kernel<<<ROSA_B * ROSA_T, 192, 0, stream>>>(idx, emb, best, out);
}